// NAA_54709293416830
// MI455X (gfx1250) — compile-verified
//
#include <hip/hip_runtime.h>
#include <stdint.h>

typedef __attribute__((ext_vector_type(2))) float v2f;
typedef __attribute__((ext_vector_type(8))) float v8f;
typedef __attribute__((ext_vector_type(4))) int   v4i;

typedef __attribute__((address_space(1))) v4i* gv4i_p;  // global int4*
typedef __attribute__((address_space(3))) v4i* sv4i_p;  // LDS int4*

#define HAVE_ASYNC_LDS  __has_builtin(__builtin_amdgcn_global_load_async_to_lds_b128)
#define HAVE_WAIT_ASYNC __has_builtin(__builtin_amdgcn_s_wait_asynccnt)
#define HAVE_WMMA_F32X4 __has_builtin(__builtin_amdgcn_wmma_f32_16x16x4_f32)

#if !HAVE_ASYNC_LDS
#warning "probe: __builtin_amdgcn_global_load_async_to_lds_b128 NOT available"
#endif
#if !HAVE_WAIT_ASYNC
#warning "probe: __builtin_amdgcn_s_wait_asynccnt NOT available"
#endif
#if !HAVE_WMMA_F32X4
#warning "probe: __builtin_amdgcn_wmma_f32_16x16x4_f32 NOT available"
#endif

// One block per (class-occurrence, output segment). 64 threads = 2 waves (wave32).
// Thread t == block-row l in [0,64).
__global__ __launch_bounds__(64) void naa_norm_store(
    const float* __restrict__ attribute,   // (2000, 1008) f32
    const float* __restrict__ betas,       // (63,) f32
    const int*   __restrict__ seencls,     // (1600,)
    const int*   __restrict__ unseencls,   // (400,)
    float*       __restrict__ out)         // zsl | seen | gzsl, concat flat
{
  __shared__ __align__(16) float sA[1008]; // attribute row
  __shared__ float sG[64];                 // per-group sum of squares (63 used)
  __shared__ float sBase;                  // total sum of squares

  const int b = (int)blockIdx.x;
  const int t = (int)threadIdx.x;

  int cls;
  size_t outbase, tot, col0;
  if (b < 400) {                            // attribute_zsl: (1008, 25600)
    cls = unseencls[b];
    outbase = 0ull;         tot = 25600;  col0 = (size_t)b * 64;
  } else if (b < 2000) {                    // attribute_seen: (1008, 102400)
    const int j = b - 400;
    cls = seencls[j];
    outbase = 25804800ull;  tot = 102400; col0 = (size_t)j * 64;
  } else {                                  // attribute_gzsl: (1008, 128000)
    const int j = b - 2000;
    cls = j;
    outbase = 129024000ull; tot = 128000; col0 = (size_t)j * 64;
  }

  const float* __restrict__ arow = attribute + (size_t)cls * 1008;

  // ---- stage the 4032-byte attribute row into LDS ----
  // 252 b128 transfers; async DMA to LDS on CDNA5 (tracked by ASYNCcnt).
#if HAVE_ASYNC_LDS
  #pragma unroll
  for (int i = 0; i < 4; ++i) {
    const int idx = t + 64 * i;
    if (idx < 252) {
      __builtin_amdgcn_global_load_async_to_lds_b128(
          (gv4i_p)(arow + idx * 4),
          (sv4i_p)(sA + idx * 4),
          /*imm offset*/0, /*cpol*/0);
    }
  }
#if HAVE_WAIT_ASYNC
  __builtin_amdgcn_s_wait_asynccnt(0);
#else
  asm volatile("s_wait_asynccnt 0" ::: "memory");
#endif
#else
  #pragma unroll
  for (int i = 0; i < 4; ++i) {
    const int idx = t + 64 * i;
    if (idx < 252) {
      ((float4*)sA)[idx] = ((const float4*)arow)[idx];
    }
  }
#endif
  __syncthreads();

  // ---- 63 group sums-of-squares: groupsq[g] = sum_{k<16} sA[16g+k]^2 ----
#if HAVE_WMMA_F32X4
  // D(16x16) = A(16x4) x B(4x16) + C, exact f32 math.
  // A[m,k] = sA[16*(gbase+m) + 4*chunk + k]^2 ; B = all-ones, so every
  // column of D holds groupsq. A layout (ISA 7.12.2): lane L -> M = L&15,
  // VGPR pair holds K={0,1} (lanes 0-15) or K={2,3} (lanes 16-31).
  if (t < 32) {
    const int lane  = t;
    const int m     = lane & 15;
    const int khalf = (lane >> 4) << 1;   // 0 or 2
    const v2f bones = (v2f){1.0f, 1.0f};
    #pragma unroll
    for (int gt = 0; gt < 4; ++gt) {
      const int g = gt * 16 + m;
      v8f acc = (v8f){0.f, 0.f, 0.f, 0.f, 0.f, 0.f, 0.f, 0.f};
      #pragma unroll
      for (int chunk = 0; chunk < 4; ++chunk) {
        v2f a;
        if (g < 63) {
          const float s0 = sA[g * 16 + chunk * 4 + khalf];
          const float s1 = sA[g * 16 + chunk * 4 + khalf + 1];
          a = (v2f){s0 * s0, s1 * s1};
        } else {
          a = (v2f){0.f, 0.f};
        }
        acc = __builtin_amdgcn_wmma_f32_16x16x4_f32(
            /*neg_a=*/false, a, /*neg_b=*/false, bones,
            /*c_mod=*/(short)0, acc, /*reuse_a=*/false, /*reuse_b=*/false);
      }
      // C/D layout: lane L, VGPR v -> row M = v + 8*(L>=16), all N identical.
      if (m == 0) {
        const int rbase = gt * 16 + ((lane >> 4) ? 8 : 0);
        #pragma unroll
        for (int v = 0; v < 8; ++v) {
          const int g2 = rbase + v;
          if (g2 < 63) sG[g2] = acc[v];
        }
      }
    }
  }
#else
  if (t < 63) {
    float s = 0.f;
    #pragma unroll
    for (int k = 0; k < 16; ++k) {
      const float x = sA[t * 16 + k];
      s += x * x;
    }
    sG[t] = s;
  }
#endif
  __syncthreads();

  if (t == 0) {
    float s = 0.f;
    for (int g = 0; g < 63; ++g) s += sG[g];
    sBase = s;
  }
  __syncthreads();

  // ---- per-row inverse norm (analytic) ----
  const int l = t;
  float beta = 0.0f;
  int   glo  = 1 << 30;            // sentinel: no replaced group for l<2
  float ss   = sBase;
  if (l >= 2) {
    beta = betas[l - 2];
    ss   = ss - sG[l - 1] + 16.0f * beta * beta;
    glo  = (l - 1) * 16;
  }
  const float nrm        = sqrtf(ss);
  const float inv        = 1.0f / fmaxf(nrm, 1e-12f);
  const float betascaled = beta * inv;

  // ---- transposed streaming stores: out[c, col0+l] ----
  // 32 lanes x b32 = one fully coalesced 128B line per wave per c.
  float* __restrict__ ocol = out + outbase + col0 + (size_t)l;
  #pragma unroll 8
  for (int c = 0; c < 1008; ++c) {
    float v = sA[c] * inv;
    if ((unsigned)(c - glo) < 16u) v = betascaled;  // replaced group columns
    __builtin_nontemporal_store(v, ocol + (size_t)c * tot);
  }
}

extern "C" void kernel_launch(void* const* d_in, const int* in_sizes, int n_in,
                              void* d_out, int out_size, void* d_ws, size_t ws_size,
                              hipStream_t stream) {
  (void)in_sizes; (void)n_in; (void)out_size; (void)d_ws; (void)ws_size;
  const float* attribute = (const float*)d_in[0];
  const float* betas     = (const float*)d_in[1];
  // d_in[2] = group_cols (arange-structured; not needed)
  const int*   seencls   = (const int*)d_in[3];
  const int*   unseencls = (const int*)d_in[4];
  float*       out       = (float*)d_out;

  // 400 (zsl) + 1600 (seen) + 2000 (gzsl) blocks, one class-row each.
  naa_norm_store<<<dim3(4000), dim3(64), 0, stream>>>(
      attribute, betas, seencls, unseencls, out);
}